// Retrive_at_k_15573551415403
// MI455X (gfx1250) — compile-verified
//
#include <hip/hip_runtime.h>

// f32 WMMA fragment types (CDNA5: V_WMMA_F32_16X16X4_F32)
typedef __attribute__((ext_vector_type(2))) float v2f;  // A or B fragment (2 VGPRs)
typedef __attribute__((ext_vector_type(8))) float v8f;  // C/D accumulator (8 VGPRs)

#define DIM    128          // feature dim (reference: D=128)
#define KSTEPS (DIM / 4)    // 32 WMMA k-steps of K=4
#define TOPK   10           // retrieval cutoff
#define NG     5            // groundtruth indices per row
#define WAVES  8            // waves per block
#define BLOCK  (WAVES * 32)

__global__ void zero_out_kernel(float* out) { out[0] = 0.0f; }

__global__ __launch_bounds__(BLOCK)
void retrieval_at_k_kernel(const float* __restrict__ A,        // [N, 128] modality1
                           const float* __restrict__ Bm,       // [M, 128] modality2
                           const long long* __restrict__ gt,   // [N, NG] int64
                           float* __restrict__ out,
                           int M, float invN)
{
    __shared__ float    lds_tile[WAVES][16 * 16];
    __shared__ unsigned lds_cnt[WAVES][16][NG];

    const int lane    = threadIdx.x & 31;
    const int wave    = threadIdx.x >> 5;
    const int rowbase = (blockIdx.x * WAVES + wave) * 16;

    const int lrow = lane & 15;   // column-in-tile (B/C) or row (A), per WMMA layout
    const int hi   = lane >> 4;   // half-wave selector -> K offset of +2

    // ---- Preload A fragments for this wave's 16 rows: 16x128 f32 in 64 VGPRs ----
    const float* arow = A + (size_t)(rowbase + lrow) * DIM + 2 * hi;
    v2f av[KSTEPS];
    #pragma unroll
    for (int kk = 0; kk < KSTEPS; ++kk)
        av[kk] = *(const v2f*)(arow + kk * 4);

    // ---- Phase 1: thresholds t[v][g] = score(row, gt[row][g]) via identical WMMA chain ----
    float thr[8][NG];
    const long long* gtrow = gt + (size_t)(rowbase + lrow) * NG;
    #pragma unroll
    for (int g = 0; g < NG; ++g) {
        const int    gcol = (int)gtrow[g];
        const float* brow = Bm + (size_t)gcol * DIM + 2 * hi;
        v8f acc = {0.f, 0.f, 0.f, 0.f, 0.f, 0.f, 0.f, 0.f};
        #pragma unroll
        for (int kk = 0; kk < KSTEPS; ++kk) {
            v2f bv = *(const v2f*)(brow + kk * 4);
            acc = __builtin_amdgcn_wmma_f32_16x16x4_f32(false, av[kk], false, bv,
                                                        (short)0, acc, false, false);
        }
        // Column c of this gather-tile is feat2[gt[rowbase+c][g]]; threshold for
        // row c is the diagonal element (c, c). Spill tile to LDS, read diagonal.
        #pragma unroll
        for (int v = 0; v < 8; ++v)
            lds_tile[wave][(v + 8 * hi) * 16 + lrow] = acc[v];
        __syncthreads();
        #pragma unroll
        for (int v = 0; v < 8; ++v) {
            const int r = v + 8 * hi;
            thr[v][g] = lds_tile[wave][r * 16 + r];
        }
        __syncthreads();
    }

    // ---- Phase 2: full GEMM over all M columns, counting scores > threshold ----
    unsigned cnt[8][NG];
    #pragma unroll
    for (int v = 0; v < 8; ++v)
        #pragma unroll
        for (int g = 0; g < NG; ++g)
            cnt[v][g] = 0u;

    for (int colbase = 0; colbase < M; colbase += 16) {
        const float* bptr = Bm + (size_t)(colbase + lrow) * DIM + 2 * hi;
        v8f acc = {0.f, 0.f, 0.f, 0.f, 0.f, 0.f, 0.f, 0.f};
        #pragma unroll
        for (int kk = 0; kk < KSTEPS; ++kk) {
            v2f bv = *(const v2f*)(bptr + kk * 4);
            acc = __builtin_amdgcn_wmma_f32_16x16x4_f32(false, av[kk], false, bv,
                                                        (short)0, acc, false, false);
        }
        // Strict '>' excludes the gt column itself (bitwise-equal to threshold).
        #pragma unroll
        for (int v = 0; v < 8; ++v) {
            const float s = acc[v];
            #pragma unroll
            for (int g = 0; g < NG; ++g)
                cnt[v][g] += (s > thr[v][g]) ? 1u : 0u;
        }
    }

    // ---- Reduce counts across the 16 lanes holding each row (LDS atomics) ----
    for (int i = (int)lane; i < 16 * NG; i += 32)
        lds_cnt[wave][i / NG][i % NG] = 0u;
    __syncthreads();
    #pragma unroll
    for (int v = 0; v < 8; ++v)
        #pragma unroll
        for (int g = 0; g < NG; ++g)
            atomicAdd(&lds_cnt[wave][v + 8 * hi][g], cnt[v][g]);
    __syncthreads();

    // ---- Row success: any g with fewer than TOPK strictly-greater scores ----
    if (lane < 16) {
        unsigned mn = lds_cnt[wave][lane][0];
        #pragma unroll
        for (int g = 1; g < NG; ++g) {
            unsigned c = lds_cnt[wave][lane][g];
            mn = (c < mn) ? c : mn;
        }
        if (mn < TOPK)
            atomicAdd(out, invN);  // invN = 2^-13: every partial sum k/8192 is exact
    }
}

extern "C" void kernel_launch(void* const* d_in, const int* in_sizes, int n_in,
                              void* d_out, int out_size, void* d_ws, size_t ws_size,
                              hipStream_t stream) {
    const float*     A   = (const float*)d_in[0];
    const float*     Bm  = (const float*)d_in[1];
    const long long* gt  = (const long long*)d_in[2];
    float*           out = (float*)d_out;

    const int N = in_sizes[0] / DIM;  // 8192
    const int M = in_sizes[1] / DIM;  // 8192

    zero_out_kernel<<<1, 1, 0, stream>>>(out);

    const int blocks = N / (16 * WAVES);  // 64 blocks x 256 threads
    retrieval_at_k_kernel<<<blocks, BLOCK, 0, stream>>>(A, Bm, gt, out, M, 1.0f / (float)N);
}